// OnlineTripletLoss_82394652607216
// MI455X (gfx1250) — compile-verified
//
#include <hip/hip_runtime.h>

// CDNA5 / gfx1250: wave32, WMMA F32 16x16x4 (fp32 in, fp32 acc).
typedef float v2f __attribute__((ext_vector_type(2)));
typedef float v8f __attribute__((ext_vector_type(8)));

#define D_DIM 512
#define MARGIN_INV (1.0f / 0.7f)

// Zero the 3 accumulators in workspace (harness poisons ws once; we must
// re-zero every call so graph replays are deterministic).
__global__ void triplet_init_kernel(float* __restrict__ acc) {
    if (threadIdx.x < 3) acc[threadIdx.x] = 0.0f;
}

// One wave handles 16 triplets per group iteration:
//   A (16x4 fp32)  = 16 gathered anchor rows, K-chunk of 4
//   B (4x16 fp32)  = 16 gathered pos/neg rows (transposed layout)
//   C accumulates the 16x16 product; diag(C) = the 16 dot products.
// fp32 A-operand layout: lane L<16 holds row M=L, K = k0+0..1 (2 VGPRs);
// lane L>=16 holds row M=L-16, K = k0+2..3. B mirrors with N=column.
// => per lane: one aligned float2 load per matrix per K-step.
__global__ void __launch_bounds__(256)
triplet_wmma_kernel(const float* __restrict__ emb,
                    const int*   __restrict__ trip,   // [T][3] indices
                    float*       __restrict__ acc,    // ws: [e_sum, (1-ap)_sum, (1-an)_sum]
                    int numGroups)                    // T / 16
{
    const int lane          = threadIdx.x & 31;
    const int waveInBlock   = threadIdx.x >> 5;
    const int wavesPerBlock = blockDim.x >> 5;
    const int globalWave    = blockIdx.x * wavesPerBlock + waveInBlock;
    const int totalWaves    = gridDim.x * wavesPerBlock;

    const int m    = lane & 15;          // triplet-in-group / matrix row (or col)
    const int koff = (lane >> 4) << 1;   // 0 for lanes 0-15, 2 for lanes 16-31

    float part_e = 0.0f, part_s1 = 0.0f, part_s2 = 0.0f;

    for (int g = globalWave; g < numGroups; g += totalWaves) {  // wave-uniform loop
        const long long t  = (long long)g * 16 + m;
        const int ia = trip[3 * t + 0];
        const int ip = trip[3 * t + 1];
        const int in_ = trip[3 * t + 2];

        const float* __restrict__ rowA = emb + (long long)ia * D_DIM + koff;
        const float* __restrict__ rowP = emb + (long long)ip * D_DIM + koff;
        const float* __restrict__ rowN = emb + (long long)in_ * D_DIM + koff;

        v8f cap = {};   // accumulates anchor . positive tile
        v8f can = {};   // accumulates anchor . negative tile

#pragma unroll 8
        for (int k0 = 0; k0 < D_DIM; k0 += 4) {
            v2f a = *(const v2f*)(rowA + k0);   // global_load_b64 gathers (L2 resident)
            v2f p = *(const v2f*)(rowP + k0);
            v2f n = *(const v2f*)(rowN + k0);
            // D = A x B + C  (fp32, 16x16x4)
            cap = __builtin_amdgcn_wmma_f32_16x16x4_f32(
                false, a, false, p, (short)0, cap, false, false);
            can = __builtin_amdgcn_wmma_f32_16x16x4_f32(
                false, a, false, n, (short)0, can, false, false);
        }

        // Diagonal extraction from 16x16 fp32 C layout:
        //   i in 0..7  -> lane i,    VGPR i
        //   i in 8..15 -> lane 16+i, VGPR i-8
        const bool valid = (lane < 8) || (lane >= 24);
        const int  v     = (lane < 8) ? lane : (lane - 24);
        float ap = cap[0];
        float an = can[0];
#pragma unroll
        for (int j = 1; j < 8; ++j) {
            ap = (v == j) ? cap[j] : ap;
            an = (v == j) ? can[j] : an;
        }

        ap = fminf(fmaxf(ap, -1.0f), 1.0f);
        an = fminf(fmaxf(an, -1.0f), 1.0f);
        const float e = __expf((an - ap) * MARGIN_INV);

        part_e  += valid ? e            : 0.0f;
        part_s1 += valid ? (1.0f - ap)  : 0.0f;
        part_s2 += valid ? (1.0f - an)  : 0.0f;
    }

    // wave32 butterfly reduction
#pragma unroll
    for (int off = 16; off > 0; off >>= 1) {
        part_e  += __shfl_xor(part_e,  off, 32);
        part_s1 += __shfl_xor(part_s1, off, 32);
        part_s2 += __shfl_xor(part_s2, off, 32);
    }

    if (lane == 0) {
        atomicAdd(&acc[0], part_e);
        atomicAdd(&acc[1], part_s1);
        atomicAdd(&acc[2], part_s2);
    }
}

__global__ void triplet_finalize_kernel(const float* __restrict__ acc,
                                        float* __restrict__ out,
                                        float invT) {
    if (threadIdx.x < 3) out[threadIdx.x] = acc[threadIdx.x] * invT;
}

extern "C" void kernel_launch(void* const* d_in, const int* in_sizes, int n_in,
                              void* d_out, int out_size, void* d_ws, size_t ws_size,
                              hipStream_t stream) {
    const float* emb  = (const float*)d_in[0];
    const int*   trip = (const int*)d_in[1];
    float* acc = (float*)d_ws;
    float* out = (float*)d_out;

    const int T = in_sizes[1] / 3;        // 131072
    const int numGroups = T / 16;         // 8192 groups of 16 triplets

    triplet_init_kernel<<<1, 64, 0, stream>>>(acc);

    const int block = 256;                // 8 wave32s per workgroup
    const int grid  = 256;                // 2048 waves -> 4 groups/wave
    triplet_wmma_kernel<<<grid, block, 0, stream>>>(emb, trip, acc, numGroups);

    triplet_finalize_kernel<<<1, 64, 0, stream>>>(acc, out, 1.0f / (float)T);
}